// SortLayer_56281251447202
// MI455X (gfx1250) — compile-verified
//
#include <hip/hip_runtime.h>
#include <stdint.h>

// Problem constants (match reference)
#define BB 128      // batch
#define SS 1024     // sequence
#define ZZ 128      // feature
#define ADD 2
#define NCH 3       // 1 real + ADD sampled
#define DD 10
#define CERT 10.0f

// ---------------------------------------------------------------------------
// numerically-stable sigmoid, matching jax.nn.sigmoid's two-branch form
// ---------------------------------------------------------------------------
__device__ __forceinline__ float sigmoid_stable(float v) {
    if (v >= 0.0f) {
        return 1.0f / (1.0f + expf(-v));
    } else {
        float e = expf(v);
        return e / (1.0f + e);
    }
}

// ---------------------------------------------------------------------------
// Kernel 1: segmented bitonic sort of one full key row (1024 floats) in LDS,
// sorting every aligned block of c = S>>d ascending, then emit the median of
// each block. One workgroup per batch row for EVERY depth (grid = B), which
// avoids the 65k near-empty dispatches a block-per-WG scheme needs at d=9.
// Correctness of the segmented network: after stages k=2..c/2 with the
// standard alternating direction ((i&k)==0), every aligned c-segment is
// ascending-then-descending (bitonic); the final k==c stage with direction
// forced ascending therefore sorts every c-segment ascending.
// ---------------------------------------------------------------------------
__global__ void pivot_kernel(const float* __restrict__ keys,
                             float* __restrict__ piv, int d) {
    __shared__ float lds[SS];
    const int c  = SS >> d;
    const int nb = 1 << d;
    const int b  = blockIdx.x;
    for (int i = threadIdx.x; i < SS; i += blockDim.x)
        lds[i] = keys[(size_t)b * SS + i];
    __syncthreads();
    for (int k = 2; k <= c; k <<= 1) {
        for (int j = k >> 1; j > 0; j >>= 1) {
            for (int i = threadIdx.x; i < SS; i += blockDim.x) {
                int ixj = i ^ j;
                if (ixj > i) {
                    bool up = (k == c) ? true : ((i & k) == 0);
                    float a = lds[i], bv = lds[ixj];
                    if (up ? (a > bv) : (a < bv)) { lds[i] = bv; lds[ixj] = a; }
                }
            }
            __syncthreads();
        }
    }
    for (int blk = threadIdx.x; blk < nb; blk += blockDim.x) {
        int bs = blk * c;
        piv[b * nb + blk] = 0.5f * (lds[bs + c / 2 - 1] + lds[bs + c / 2]);
    }
}

// ---------------------------------------------------------------------------
// Kernel 2: per-(b,n) row: choice bit, stable-partition index via inclusive
// scan (LDS, Hillis-Steele over 1024), inverse permutation, and raw prob
// product (plain f32 product, matching reference underflow behavior).
// Launch: grid = B*NCH, block = 1024.
// ---------------------------------------------------------------------------
__global__ void probs_idx_kernel(const float* __restrict__ keys,
                                 const float* __restrict__ piv,
                                 const int* __restrict__ sampled,
                                 int* __restrict__ idx,
                                 int* __restrict__ inv,
                                 float* __restrict__ praw, int d) {
    __shared__ int   scan[SS];
    __shared__ float prod[SS];
    const int s = threadIdx.x;          // 0..1023
    const int b = blockIdx.x / NCH;
    const int n = blockIdx.x % NCH;
    const int c = SS >> d;
    const int nb = 1 << d;
    const int blk  = s / c;
    const int bs   = blk * c;
    const int half = c >> 1;

    float key = keys[(size_t)b * SS + s];
    float pv  = piv[b * nb + blk];
    float off = sigmoid_stable((key - pv) * CERT);
    int ch;
    if (n == 0) {
        ch = (off > 0.5f) ? 1 : 0;
    } else {
        ch = (sampled[(((size_t)d * BB + b) * ADD + (n - 1)) * SS + s] != 0) ? 1 : 0;
    }
    float p = ch ? off : (1.0f - off);

    scan[s] = ch;
    prod[s] = p;
    // inclusive scan over full row
    for (int o = 1; o < SS; o <<= 1) {
        __syncthreads();
        int t = (s >= o) ? scan[s - o] : 0;
        __syncthreads();
        scan[s] += t;
    }
    __syncthreads();
    int excl_s  = scan[s] - ch;
    int excl_bs = (bs == 0) ? 0 : scan[bs - 1];
    int ones_b  = excl_s - excl_bs;
    int pos     = s - bs;
    int zeros_b = pos - ones_b;
    int target  = bs + (ch ? (half + ones_b) : zeros_b);

    idx[((size_t)b * NCH + n) * SS + s] = target;
    int tcl = target < 0 ? 0 : (target >= SS ? SS - 1 : target);  // safety clamp
    inv[((size_t)b * NCH + n) * SS + tcl] = s;

    // tree product of p over the row
    for (int str = SS / 2; str > 0; str >>= 1) {
        __syncthreads();
        if (s < str) prod[s] *= prod[s + str];
    }
    if (s == 0) praw[b * NCH + n] = prod[0];
}

// ---------------------------------------------------------------------------
// Kernel 3: duplicate-routing detection + normalization. grid = B, block = 256.
// ---------------------------------------------------------------------------
__global__ void dup_norm_kernel(const int* __restrict__ idx,
                                const float* __restrict__ praw,
                                float* __restrict__ pfin) {
    __shared__ int eq[3];   // eq10, eq20, eq21
    const int b = blockIdx.x;
    if (threadIdx.x < 3) eq[threadIdx.x] = 1;
    __syncthreads();
    const int* r0 = idx + (size_t)b * NCH * SS;
    const int* r1 = r0 + SS;
    const int* r2 = r1 + SS;
    for (int s = threadIdx.x; s < SS; s += blockDim.x) {
        int a = r0[s], v1 = r1[s], v2 = r2[s];
        if (v1 != a)  eq[0] = 0;   // benign race: only 0 written
        if (v2 != a)  eq[1] = 0;
        if (v2 != v1) eq[2] = 0;
    }
    __syncthreads();
    if (threadIdx.x == 0) {
        float p0 = praw[b * NCH + 0];
        float p1 = eq[0] ? 0.0f : praw[b * NCH + 1];
        float p2 = (eq[1] || eq[2]) ? 0.0f : praw[b * NCH + 2];
        float ssum = p0 + p1 + p2;
        pfin[b * NCH + 0] = p0 / ssum;
        pfin[b * NCH + 1] = p1 / ssum;
        pfin[b * NCH + 2] = p2 / ssum;
    }
}

// ---------------------------------------------------------------------------
// Kernel 4: gather/update. One wave32 per output row (Z=128 floats = 32
// lanes x float4). The 3 source rows are staged into LDS with CDNA5 async
// global->LDS B128 transfers, synchronized with s_wait_asynccnt, then
// combined with the normalized routing probabilities. The scalar keys
// combine is issued before the async wait so both transfers overlap.
// Launch: grid = B*S/8, block = 256 (8 waves, 8 rows per WG).
// ---------------------------------------------------------------------------
#define ROWS_PER_WG 8
__global__ void gather_kernel(const float* __restrict__ xsrc,
                              const float* __restrict__ ksrc,
                              const int* __restrict__ inv,
                              const float* __restrict__ pfin,
                              float* __restrict__ xdst,
                              float* __restrict__ kdst) {
    __shared__ __align__(16) float tile[ROWS_PER_WG * NCH * ZZ];  // 12 KB
    const int lane = threadIdx.x & 31;
    const int wave = threadIdx.x >> 5;
    const long long row = (long long)blockIdx.x * ROWS_PER_WG + wave;
    const int b = (int)(row / SS);
    const int t = (int)(row % SS);

    const int i0 = inv[((size_t)b * NCH + 0) * SS + t];
    const int i1 = inv[((size_t)b * NCH + 1) * SS + t];
    const int i2 = inv[((size_t)b * NCH + 2) * SS + t];
    const float p0 = pfin[b * NCH + 0];
    const float p1 = pfin[b * NCH + 1];
    const float p2 = pfin[b * NCH + 2];

    const float* base = xsrc + (size_t)b * SS * ZZ;
    // LDS byte offset of this wave's staging area (low 32 bits of the flat
    // address of a __shared__ object are the group-relative LDS offset)
    unsigned lbase = (unsigned)(size_t)(void*)&tile[wave * NCH * ZZ];
    const char* g0 = (const char*)(base + (size_t)i0 * ZZ) + lane * 16;
    const char* g1 = (const char*)(base + (size_t)i1 * ZZ) + lane * 16;
    const char* g2 = (const char*)(base + (size_t)i2 * ZZ) + lane * 16;
    unsigned l0 = lbase + 0 * ZZ * 4 + lane * 16;
    unsigned l1 = lbase + 1 * ZZ * 4 + lane * 16;
    unsigned l2 = lbase + 2 * ZZ * 4 + lane * 16;

    asm volatile("global_load_async_to_lds_b128 %0, %1, off"
                 :: "v"(l0), "v"(g0) : "memory");
    asm volatile("global_load_async_to_lds_b128 %0, %1, off"
                 :: "v"(l1), "v"(g1) : "memory");
    asm volatile("global_load_async_to_lds_b128 %0, %1, off"
                 :: "v"(l2), "v"(g2) : "memory");

    // keys combine: issue the scalar loads while the async x transfers fly
    if (lane == 0) {
        const float* kb = ksrc + (size_t)b * SS;
        kdst[(size_t)b * SS + t] = p0 * kb[i0] + p1 * kb[i1] + p2 * kb[i2];
    }

    asm volatile("s_wait_asynccnt 0" ::: "memory");

    const float4* t4 = (const float4*)&tile[wave * NCH * ZZ];
    float4 a  = t4[0 * 32 + lane];
    float4 c1 = t4[1 * 32 + lane];
    float4 c2 = t4[2 * 32 + lane];
    float4 o;
    o.x = p0 * a.x + p1 * c1.x + p2 * c2.x;
    o.y = p0 * a.y + p1 * c1.y + p2 * c2.y;
    o.z = p0 * a.z + p1 * c1.z + p2 * c2.z;
    o.w = p0 * a.w + p1 * c1.w + p2 * c2.w;
    ((float4*)(xdst + ((size_t)b * SS + t) * ZZ))[lane] = o;
}

// ---------------------------------------------------------------------------
// Host launcher. Workspace layout (bytes):
//   [0, 64MB)      ping x buffer
//   +64MB          keys ping A (512 KB)
//   ...            keys ping B, pivots, idx, inv, praw, pfin  (~68.3 MB total)
// x ping-pongs between ws and d_out's x-region so depth 9 writes d_out.
// ---------------------------------------------------------------------------
extern "C" void kernel_launch(void* const* d_in, const int* in_sizes, int n_in,
                              void* d_out, int out_size, void* d_ws, size_t ws_size,
                              hipStream_t stream) {
    (void)in_sizes; (void)n_in; (void)out_size; (void)ws_size;
    const float* in_x = (const float*)d_in[0];
    const float* in_k = (const float*)d_in[1];
    const int*   samp = (const int*)d_in[2];

    float* out_x = (float*)d_out;
    float* out_k = out_x + (size_t)BB * SS * ZZ;

    char* ws = (char*)d_ws;
    float* ws_x = (float*)(ws + 0);                       // 67108864 B
    float* kA   = (float*)(ws + 67108864);                //   524288 B
    float* kB   = (float*)(ws + 67633152);                //   524288 B
    float* piv  = (float*)(ws + 68157440);                //   262144 B
    int*   idx  = (int*)  (ws + 68419584);                //  1572864 B
    int*   inv  = (int*)  (ws + 69992448);                //  1572864 B
    float* praw = (float*)(ws + 71565312);                //     1536 B
    float* pfin = (float*)(ws + 71566848);                //     1536 B

    for (int d = 0; d < DD; ++d) {
        const float* xsrc = (d == 0) ? in_x : ((d & 1) ? ws_x : out_x);
        float*       xdst = (d & 1) ? out_x : ws_x;            // d=9 -> out_x
        const float* ksrc = (d == 0) ? in_k : ((d & 1) ? kA : kB);
        float*       kdst = (d == DD - 1) ? out_k : ((d & 1) ? kB : kA);

        pivot_kernel<<<BB, 256, 0, stream>>>(ksrc, piv, d);
        probs_idx_kernel<<<BB * NCH, 1024, 0, stream>>>(ksrc, piv, samp,
                                                        idx, inv, praw, d);
        dup_norm_kernel<<<BB, 256, 0, stream>>>(idx, praw, pfin);
        gather_kernel<<<(BB * SS) / ROWS_PER_WG, 256, 0, stream>>>(
            xsrc, ksrc, inv, pfin, xdst, kdst);
    }
}